// VGAE_encoder_45904610460272
// MI455X (gfx1250) — compile-verified
//
#include <hip/hip_runtime.h>

typedef __attribute__((ext_vector_type(2))) float v2f;
typedef __attribute__((ext_vector_type(8))) float v8f;

// ---------------------------------------------------------------- utilities

__global__ __launch_bounds__(256) void k_zero(float* __restrict__ deg,
                                              float* __restrict__ agg,
                                              int n, int n32) {
  int i = blockIdx.x * 256 + threadIdx.x;
  if (i < n)   deg[i] = 0.0f;
  if (i < n32) agg[i] = 0.0f;
}

__global__ __launch_bounds__(256) void k_deg(const int* __restrict__ col,
                                             float* __restrict__ deg, int E) {
  int e = blockIdx.x * 256 + threadIdx.x;
  if (e < E) atomicAdd(&deg[col[e]], 1.0f);
}

__global__ __launch_bounds__(256) void k_rsqrt(float* __restrict__ deg_dis, int n) {
  int i = blockIdx.x * 256 + threadIdx.x;
  if (i < n) deg_dis[i] = rsqrtf(deg_dis[i] + 1.0f);  // +1 self-loop
}

// ------------------------------------------------- layer 1 GEMM: h = x @ W1
// fp32 WMMA 16x16x4. One wave per 16-row tile, full 32-wide output.
// A frag (16x4 f32): lanes 0-15 hold row M=lane, V0=K0,V1=K1;
//                    lanes 16-31 hold row M=lane-16, V0=K2,V1=K3.
// B frag (4x16 f32): mirrored (lane = column, K pairs split across halves).
// C/D (16x16 f32):   VGPR r: M=r (lanes 0-15), M=r+8 (lanes 16-31), N=lane&15.
__global__ __launch_bounds__(256) void k_gemm1(const float* __restrict__ x,
                                               const float* __restrict__ W1,
                                               float* __restrict__ h, int n) {
  __shared__ float sW[256 * 32];  // 32 KB of the WGP's 320 KB LDS
  int tid = threadIdx.x;
  for (int j = tid; j < 256 * 32; j += 256) sW[j] = W1[j];
  __syncthreads();

  int wave = tid >> 5, lane = tid & 31;
  int mtile = blockIdx.x * 8 + wave;
  if (mtile >= (n >> 4)) return;

  int mbase = mtile << 4;
  int mrow  = mbase + (lane & 15);
  int kg    = lane >> 4;       // 0: K0/K1,  1: K2/K3
  int nloc  = lane & 15;

  v8f c0 = {}; v8f c1 = {};
  const float* xr = x + (size_t)mrow * 256;

  for (int k = 0; k < 256; k += 4) {
    int kk = k + 2 * kg;
    v2f a = *(const v2f*)(xr + kk);           // 8B-aligned (kk even)
    v2f b0, b1v;
    b0.x  = sW[kk * 32 + nloc];       b0.y  = sW[(kk + 1) * 32 + nloc];
    b1v.x = sW[kk * 32 + 16 + nloc];  b1v.y = sW[(kk + 1) * 32 + 16 + nloc];
    c0 = __builtin_amdgcn_wmma_f32_16x16x4_f32(false, a, false, b0,  (short)0, c0, false, false);
    c1 = __builtin_amdgcn_wmma_f32_16x16x4_f32(false, a, false, b1v, (short)0, c1, false, false);
  }

  float* hr = h + (size_t)(mbase + 8 * kg) * 32 + nloc;
  #pragma unroll
  for (int r = 0; r < 8; ++r) {
    hr[(size_t)r * 32]      = c0[r];
    hr[(size_t)r * 32 + 16] = c1[r];
  }
}

// ------------------------------- edge aggregation: agg[col] += src[row]*norm
// 8 threads per edge, float4 per thread; L2-resident atomics.
__global__ __launch_bounds__(256) void k_edge_agg(const int* __restrict__ row,
                                                  const int* __restrict__ col,
                                                  const float* __restrict__ dis,
                                                  const float* __restrict__ src,
                                                  float* __restrict__ dst, int E) {
  int idx = blockIdx.x * 256 + threadIdx.x;
  int e = idx >> 3;
  if (e >= E) return;
  int c = (idx & 7) * 4;
  int r = row[e], t = col[e];
  float norm = dis[r] * dis[t];
  float4 v = *(const float4*)(src + (size_t)r * 32 + c);
  float* d = dst + (size_t)t * 32 + c;
  atomicAdd(d + 0, v.x * norm);
  atomicAdd(d + 1, v.y * norm);
  atomicAdd(d + 2, v.z * norm);
  atomicAdd(d + 3, v.w * norm);
}

// -------------------- fused: h = relu(agg + dis^2*h + b1); agg = 0 (for pass 2)
__global__ __launch_bounds__(256) void k_relu(float* __restrict__ h,
                                              float* __restrict__ agg,
                                              const float* __restrict__ dis,
                                              const float* __restrict__ b1, int n32) {
  int i = blockIdx.x * 256 + threadIdx.x;
  if (i >= n32) return;
  int node = i >> 5, f = i & 31;
  float d = dis[node];
  float v = agg[i] + d * d * h[i] + b1[f];
  h[i]   = fmaxf(v, 0.0f);
  agg[i] = 0.0f;
}

// -------- output heads: t = agg + dis^2*h ; mu = t@Wmu + bmu ; lv = t@Wlv + blv
__global__ __launch_bounds__(256) void k_out(const float* __restrict__ h,
                                             const float* __restrict__ agg,
                                             const float* __restrict__ dis,
                                             const float* __restrict__ Wmu,
                                             const float* __restrict__ bmu,
                                             const float* __restrict__ Wlv,
                                             const float* __restrict__ blv,
                                             float* __restrict__ mu,
                                             float* __restrict__ lv, int n) {
  __shared__ float sWm[32 * 16];
  __shared__ float sWl[32 * 16];
  int tid = threadIdx.x;
  for (int j = tid; j < 32 * 16; j += 256) { sWm[j] = Wmu[j]; sWl[j] = Wlv[j]; }
  __syncthreads();

  int wave = tid >> 5, lane = tid & 31;
  int mtile = blockIdx.x * 8 + wave;
  if (mtile >= (n >> 4)) return;

  int mbase = mtile << 4;
  int m     = mbase + (lane & 15);
  int kg    = lane >> 4;
  int nloc  = lane & 15;

  float dm = dis[m];
  float sn = dm * dm;

  v8f cmu = {}; v8f clv = {};
  const float* hr = h   + (size_t)m * 32;
  const float* ar = agg + (size_t)m * 32;

  #pragma unroll
  for (int k = 0; k < 32; k += 4) {
    int kk = k + 2 * kg;
    v2f hv = *(const v2f*)(hr + kk);
    v2f av = *(const v2f*)(ar + kk);
    v2f a;
    a.x = av.x + sn * hv.x;
    a.y = av.y + sn * hv.y;
    v2f bm, bl;
    bm.x = sWm[kk * 16 + nloc];  bm.y = sWm[(kk + 1) * 16 + nloc];
    bl.x = sWl[kk * 16 + nloc];  bl.y = sWl[(kk + 1) * 16 + nloc];
    cmu = __builtin_amdgcn_wmma_f32_16x16x4_f32(false, a, false, bm, (short)0, cmu, false, false);
    clv = __builtin_amdgcn_wmma_f32_16x16x4_f32(false, a, false, bl, (short)0, clv, false, false);
  }

  float bmn = bmu[nloc], bln = blv[nloc];
  float* mo = mu + (size_t)(mbase + 8 * kg) * 16 + nloc;
  float* lo = lv + (size_t)(mbase + 8 * kg) * 16 + nloc;
  #pragma unroll
  for (int r = 0; r < 8; ++r) {
    mo[r * 16] = cmu[r] + bmn;
    lo[r * 16] = clv[r] + bln;
  }
}

// ---------------------------------------------------------------- launcher

extern "C" void kernel_launch(void* const* d_in, const int* in_sizes, int n_in,
                              void* d_out, int out_size, void* d_ws, size_t ws_size,
                              hipStream_t stream) {
  const float* x   = (const float*)d_in[0];
  const int*   ei  = (const int*)  d_in[1];
  const float* W1  = (const float*)d_in[2];
  const float* b1  = (const float*)d_in[3];
  const float* Wmu = (const float*)d_in[4];
  const float* bmu = (const float*)d_in[5];
  const float* Wlv = (const float*)d_in[6];
  const float* blv = (const float*)d_in[7];

  int n = in_sizes[0] / 256;   // 100000 (divisible by 16)
  int E = in_sizes[1] / 2;     // 3200000
  const int* row = ei;
  const int* col = ei + E;

  float* ws   = (float*)d_ws;
  float* dis  = ws;                          // n   (deg, then rsqrt in place)
  float* hbuf = ws + n;                      // 32n (h_lin, then h)
  float* agg  = ws + n + (size_t)32 * n;     // 32n (reused for both passes)
  int n32 = 32 * n;

  int mblocks = ((n >> 4) + 7) / 8;

  k_zero <<<(n32 + 255) / 256, 256, 0, stream>>>(dis, agg, n, n32);
  k_deg  <<<(E + 255) / 256,   256, 0, stream>>>(col, dis, E);
  k_rsqrt<<<(n + 255) / 256,   256, 0, stream>>>(dis, n);

  k_gemm1<<<mblocks, 256, 0, stream>>>(x, W1, hbuf, n);

  k_edge_agg<<<(E * 8 + 255) / 256, 256, 0, stream>>>(row, col, dis, hbuf, agg, E);
  k_relu    <<<(n32 + 255) / 256,   256, 0, stream>>>(hbuf, agg, dis, b1, n32);
  k_edge_agg<<<(E * 8 + 255) / 256, 256, 0, stream>>>(row, col, dis, hbuf, agg, E);

  float* mu = (float*)d_out;
  float* lv = (float*)d_out + (size_t)n * 16;
  k_out<<<mblocks, 256, 0, stream>>>(hbuf, agg, dis, Wmu, bmu, Wlv, blv, mu, lv, n);
}